// LSTMConditioned_67422396613275
// MI455X (gfx1250) — compile-verified
//
#include <hip/hip_runtime.h>
#include <math.h>

#define N_    128
#define T_    64
#define HID_  512
#define COMB_ 1024
#define ATT_  128
#define K_    20
#define D_    2
#define FMAP_ 64
#define L_    196     // 14*14 conv3 output positions
#define ODIM_ 121     // K + 2KD + K + 1

typedef __attribute__((ext_vector_type(2))) float v2f;
typedef __attribute__((ext_vector_type(8))) float v8f;

__device__ __forceinline__ float sigm(float x) { return 1.0f / (1.0f + __expf(-x)); }

// ======================================================================
// WMMA-f32 GEMM:  C[M,Ntot] = act( A[M,K] @ B[Ntot,K]^T + bias (+C) )
// Each wave computes TN=2 adjacent 16x16 C tiles along N, sharing one A
// fragment per K-chunk. Inner structure: per 16-wide K block, issue ALL
// 12 b64 loads into distinct fragments first, then all 8 wmmas — lets
// the scheduler form a load clause and stagger s_wait_loadcnt so memory
// latency overlaps the wmma burst (instead of load->wait0->wmma chains).
// v_wmma_f32_16x16x4_f32 keeps full f32 precision; weight bytes are the
// same at any WMMA precision (weights are f32), so f32 is free here.
// f32 A-matrix 16x4 layout: lane l holds A[m = l&15][2*(l>>4) + j] in reg j.
// f32 B-matrix 4x16 layout: lane l holds B[k = 2*(l>>4) + j][n = l&15].
// Requires K % 16 == 0 (true for all K here: 512, 80, 608, 1024, 64).
// ======================================================================
template <int ACT /*0=none,1=tanh*/, int ACC /*0=store,1=accumulate*/>
__global__ __launch_bounds__(128) void gemm_wmma_f32(
    const float* __restrict__ A, int lda,
    const float* __restrict__ B, int ldb,     // B is [Ntot, K] (weight layout)
    const float* __restrict__ bias,           // may be null
    float* __restrict__ C, int ldc,
    int M, int Ntot, int K)
{
    constexpr int TN = 2;                     // N tiles per wave
    constexpr int PF = 256;                   // prefetch lookahead (floats, 1KB)
    const int lane  = threadIdx.x & 31;
    const int wave  = threadIdx.x >> 5;
    const int r15   = lane & 15;
    const int kh    = lane >> 4;              // K half (0/1) inside a K=4 chunk
    const int mBase = blockIdx.y * 16;
    const int nBase = (blockIdx.x * 4 + wave) * 16 * TN;
    if (nBase >= Ntot) return;                // wave-uniform exit

    const float* Ap = A + (size_t)(mBase + r15) * lda + 2 * kh;

    int  nCol[TN];
    bool nOK[TN];
    const float* Bp[TN];
#pragma unroll
    for (int tn = 0; tn < TN; ++tn) {
        nCol[tn] = nBase + tn * 16 + r15;
        nOK[tn]  = nCol[tn] < Ntot;
        Bp[tn]   = B + (size_t)(nOK[tn] ? nCol[tn] : 0) * ldb + 2 * kh;
    }

    v8f acc[TN];
#pragma unroll
    for (int tn = 0; tn < TN; ++tn) acc[tn] = (v8f){};

    for (int k0 = 0; k0 < K; k0 += 16) {
        // Near-scope prefetch (locality 3 -> WGP scope: fills all cache
        // levels). WGP prefetch is non-speculative, so only issue when the
        // whole lookahead stays inside the current row (always-valid VA).
        if (k0 + PF + 18 <= K) {
            __builtin_prefetch(Ap + k0 + PF, 0, 3);
#pragma unroll
            for (int tn = 0; tn < TN; ++tn) __builtin_prefetch(Bp[tn] + k0 + PF, 0, 3);
        }

        v2f av[4];
        v2f bv[TN][4];
#pragma unroll
        for (int kk = 0; kk < 4; ++kk) {
            const int k = k0 + kk * 4;
            av[kk] = (v2f){ Ap[k], Ap[k + 1] };
#pragma unroll
            for (int tn = 0; tn < TN; ++tn)
                bv[tn][kk] = (v2f){ Bp[tn][k], Bp[tn][k + 1] };
        }
#pragma unroll
        for (int kk = 0; kk < 4; ++kk)
#pragma unroll
            for (int tn = 0; tn < TN; ++tn)
                acc[tn] = __builtin_amdgcn_wmma_f32_16x16x4_f32(
                    /*neg_a=*/false, av[kk], /*neg_b=*/false, bv[tn][kk],
                    /*c_mod=*/(short)0, acc[tn], /*reuse_a=*/false, /*reuse_b=*/false);
    }

#pragma unroll
    for (int tn = 0; tn < TN; ++tn) {
        const float bn = (bias && nOK[tn]) ? bias[nCol[tn]] : 0.0f;
#pragma unroll
        for (int r = 0; r < 8; ++r) {
            const int row = mBase + r + 8 * kh;  // C/D layout: VGPR r -> M = r + 8*kh
            float v = acc[tn][r] + bn;
            float* cp = C + (size_t)row * ldc + nCol[tn];
            if (ACC && nOK[tn]) v += *cp;
            if (ACT == 1) v = tanhf(v);
            if (nOK[tn]) *cp = v;
        }
    }
}

// ======================================================================
// Setup kernels (one-time, VALU; cheap vs the 64-step scan)
// ======================================================================

// xe[n,t,:] = tanh(xprev @ xemb^T + b), se[n,:] = tanh(start @ semb^T + b)
__global__ void embed_kernel(const float* __restrict__ x,
                             const float* __restrict__ start,
                             const float* __restrict__ xemb_w, const float* __restrict__ xemb_b,
                             const float* __restrict__ semb_w, const float* __restrict__ semb_b,
                             float* __restrict__ xe, float* __restrict__ se)
{
    int idx = blockIdx.x * blockDim.x + threadIdx.x;
    if (idx >= N_ * (T_ + 1) * 16) return;
    int n = idx / ((T_ + 1) * 16);
    int r = idx % ((T_ + 1) * 16);
    int t = r / 16, j = r % 16;
    if (t < T_) {
        float acc = xemb_b[j];
        if (t > 0) {
            acc += x[((size_t)n * T_ + (t - 1)) * D_ + 0] * xemb_w[j * D_ + 0];
            acc += x[((size_t)n * T_ + (t - 1)) * D_ + 1] * xemb_w[j * D_ + 1];
        }
        xe[((size_t)n * T_ + t) * 16 + j] = tanhf(acc);
    } else {
        float acc = semb_b[j];
        acc += start[n * D_ + 0] * semb_w[j * D_ + 0];
        acc += start[n * D_ + 1] * semb_w[j * D_ + 1];
        se[n * 16 + j] = tanhf(acc);
    }
}

__global__ void conv1_kernel(const float* __restrict__ canv,
                             const float* __restrict__ w, const float* __restrict__ b,
                             float* __restrict__ out)
{
    int idx = blockIdx.x * blockDim.x + threadIdx.x;
    if (idx >= N_ * 16 * 28 * 28) return;
    int n = idx / (16 * 784);
    int r = idx % (16 * 784);
    int oc = r / 784, p = r % 784, oy = p / 28, ox = p % 28;
    float acc = b[oc];
    for (int ky = 0; ky < 3; ++ky)
        for (int kx = 0; kx < 3; ++kx) {
            int iy = oy + ky - 1, ix = ox + kx - 1;
            if (iy >= 0 && iy < 28 && ix >= 0 && ix < 28) {
                float v = (canv[(size_t)n * 784 + iy * 28 + ix] - 0.0243f) * (1.0f / 0.1383f);
                acc += w[oc * 9 + ky * 3 + kx] * v;
            }
        }
    out[idx] = fmaxf(acc, 0.0f);
}

__global__ void conv2_kernel(const float* __restrict__ in,
                             const float* __restrict__ w, const float* __restrict__ b,
                             float* __restrict__ out)
{
    int idx = blockIdx.x * blockDim.x + threadIdx.x;
    if (idx >= N_ * 32 * 28 * 28) return;
    int n = idx / (32 * 784);
    int r = idx % (32 * 784);
    int oc = r / 784, p = r % 784, oy = p / 28, ox = p % 28;
    float acc = b[oc];
    for (int ic = 0; ic < 16; ++ic)
        for (int ky = 0; ky < 3; ++ky)
            for (int kx = 0; kx < 3; ++kx) {
                int iy = oy + ky - 1, ix = ox + kx - 1;
                if (iy >= 0 && iy < 28 && ix >= 0 && ix < 28)
                    acc += w[oc * 144 + ic * 9 + ky * 3 + kx] *
                           in[((size_t)n * 16 + ic) * 784 + iy * 28 + ix];
            }
    out[idx] = fmaxf(acc, 0.0f);
}

// conv3 (stride 2) with tanh; writes directly in a-layout: a[n][l=oy*14+ox][f=oc]
__global__ void conv3_kernel(const float* __restrict__ in,
                             const float* __restrict__ w, const float* __restrict__ b,
                             float* __restrict__ a)
{
    int idx = blockIdx.x * blockDim.x + threadIdx.x;
    if (idx >= N_ * 64 * 196) return;
    int n = idx / (64 * 196);
    int r = idx % (64 * 196);
    int oc = r / 196, p = r % 196, oy = p / 14, ox = p % 14;
    float acc = b[oc];
    for (int ic = 0; ic < 32; ++ic)
        for (int ky = 0; ky < 3; ++ky)
            for (int kx = 0; kx < 3; ++kx) {
                int iy = oy * 2 + ky - 1, ix = ox * 2 + kx - 1;
                if (iy >= 0 && iy < 28 && ix >= 0 && ix < 28)
                    acc += w[oc * 288 + ic * 9 + ky * 3 + kx] *
                           in[((size_t)n * 32 + ic) * 784 + iy * 28 + ix];
            }
    a[((size_t)n * L_ + p) * FMAP_ + oc] = tanhf(acc);
}

__global__ void amean_kernel(const float* __restrict__ a, float* __restrict__ amean)
{
    int idx = blockIdx.x * blockDim.x + threadIdx.x;
    if (idx >= N_ * FMAP_) return;
    int n = idx / FMAP_, f = idx % FMAP_;
    float acc = 0.0f;
    for (int l = 0; l < L_; ++l) acc += a[((size_t)n * L_ + l) * FMAP_ + f];
    amean[idx] = acc * (1.0f / (float)L_);
}

// hidden = tanh([amean, se] @ init_w^T + init_b); even idx -> h0, odd -> c0
__global__ void init_state_kernel(const float* __restrict__ amean, const float* __restrict__ se,
                                  const float* __restrict__ init_w, const float* __restrict__ init_b,
                                  float* __restrict__ h, float* __restrict__ c)
{
    int idx = blockIdx.x * blockDim.x + threadIdx.x;
    if (idx >= N_ * 2 * HID_) return;
    int n = idx / (2 * HID_), j = idx % (2 * HID_);
    float acc = init_b[j];
    const float* wr = init_w + (size_t)j * (FMAP_ + 16);
    for (int i = 0; i < FMAP_; ++i) acc += amean[n * FMAP_ + i] * wr[i];
    for (int i = 0; i < 16; ++i)    acc += se[n * 16 + i] * wr[FMAP_ + i];
    acc = tanhf(acc);
    if ((j & 1) == 0) h[(size_t)n * HID_ + (j >> 1)] = acc;
    else              c[(size_t)n * HID_ + (j >> 1)] = acc;
}

// ======================================================================
// Per-step kernels
// ======================================================================

// attention: e = tanh(Wa + Uh)·vw + vb ; softmax over L ; z = sigmoid(betapre)*(alpha·a)
__global__ __launch_bounds__(256) void attn_kernel(
    const float* __restrict__ Wa, const float* __restrict__ Uh,
    const float* __restrict__ vw, const float* __restrict__ vb,
    const float* __restrict__ a,  const float* __restrict__ betapre,
    float* __restrict__ z)
{
    const int n = blockIdx.x;
    const int tid = threadIdx.x;
    __shared__ float sUh[ATT_];
    __shared__ float sE[L_];
    __shared__ float sRed[256];

    if (tid < ATT_) sUh[tid] = Uh[(size_t)n * ATT_ + tid];
    __syncthreads();

    if (tid < L_) {
        const float* wr = Wa + ((size_t)n * L_ + tid) * ATT_;
        float e = vb[0];
        for (int f = 0; f < ATT_; ++f) e += tanhf(wr[f] + sUh[f]) * vw[f];
        sE[tid] = e;
    }
    __syncthreads();

    // max reduce
    sRed[tid] = (tid < L_) ? sE[tid] : -3.0e38f;
    __syncthreads();
    for (int s = 128; s > 0; s >>= 1) {
        if (tid < s) sRed[tid] = fmaxf(sRed[tid], sRed[tid + s]);
        __syncthreads();
    }
    const float mx = sRed[0];
    __syncthreads();

    float ex = (tid < L_) ? __expf(sE[tid] - mx) : 0.0f;
    if (tid < L_) sE[tid] = ex;
    sRed[tid] = ex;
    __syncthreads();
    for (int s = 128; s > 0; s >>= 1) {
        if (tid < s) sRed[tid] += sRed[tid + s];
        __syncthreads();
    }
    const float inv_sum = 1.0f / sRed[0];
    __syncthreads();

    if (tid < FMAP_) {
        float acc = 0.0f;
        for (int l = 0; l < L_; ++l)
            acc += sE[l] * a[((size_t)n * L_ + l) * FMAP_ + tid];
        const float g = sigm(betapre[(size_t)n * FMAP_ + tid]);
        z[(size_t)n * FMAP_ + tid] = acc * inv_sum * g;
    }
}

__global__ void pack_xz_kernel(const float* __restrict__ xe, const float* __restrict__ z,
                               float* __restrict__ xz, int t)
{
    int idx = blockIdx.x * blockDim.x + threadIdx.x;
    if (idx >= N_ * 80) return;
    int n = idx / 80, j = idx % 80;
    xz[idx] = (j < 16) ? xe[((size_t)n * T_ + t) * 16 + j]
                       : z[(size_t)n * FMAP_ + (j - 16)];
}

// LSTM pointwise + pack u = [h, z, xt, se] (608)
__global__ void lstm_kernel(const float* __restrict__ G, float* __restrict__ c,
                            float* __restrict__ h, const float* __restrict__ z,
                            const float* __restrict__ xe, const float* __restrict__ se,
                            float* __restrict__ u, int t)
{
    int idx = blockIdx.x * blockDim.x + threadIdx.x;
    if (idx >= N_ * 608) return;
    int n = idx / 608, j = idx % 608;
    if (j < HID_) {
        const float* g = G + (size_t)n * 4 * HID_;
        float gi = g[j], gf = g[HID_ + j], gg = g[2 * HID_ + j], go = g[3 * HID_ + j];
        float cn = sigm(gf) * c[(size_t)n * HID_ + j] + sigm(gi) * tanhf(gg);
        float hn = sigm(go) * tanhf(cn);
        c[(size_t)n * HID_ + j] = cn;
        h[(size_t)n * HID_ + j] = hn;
        u[idx] = hn;
    } else if (j < HID_ + FMAP_) {
        u[idx] = z[(size_t)n * FMAP_ + (j - HID_)];
    } else if (j < HID_ + FMAP_ + 16) {
        u[idx] = xe[((size_t)n * T_ + t) * 16 + (j - HID_ - FMAP_)];
    } else {
        u[idx] = se[n * 16 + (j - HID_ - FMAP_ - 16)];
    }
}

// Output head transforms, writing the 5 concatenated output tensors.
__global__ void finalize_kernel(const float* __restrict__ ys, float* __restrict__ out)
{
    int idx = blockIdx.x * blockDim.x + threadIdx.x;
    if (idx >= N_ * T_) return;
    int n = idx / T_, t = idx % T_;
    const float* y = ys + ((size_t)t * N_ + n) * 128;  // ys layout: [T][N][128]
    const size_t NT = (size_t)N_ * T_;
    float* o_mix    = out;
    float* o_means  = out + NT * K_;
    float* o_scales = out + NT * K_ + NT * K_ * D_;
    float* o_corrs  = out + NT * K_ + 2 * NT * K_ * D_;
    float* o_vlog   = out + 2 * NT * K_ + 2 * NT * K_ * D_;
    const size_t nt = (size_t)n * T_ + t;

    float mx = -3.0e38f;
    for (int k = 0; k < K_; ++k) mx = fmaxf(mx, y[k]);
    float sm = 0.0f;
    for (int k = 0; k < K_; ++k) sm += __expf(y[k] - mx);
    float inv = 1.0f / sm;
    for (int k = 0; k < K_; ++k)          o_mix[nt * K_ + k]          = __expf(y[k] - mx) * inv;
    for (int i = 0; i < K_ * D_; ++i)     o_means[nt * K_ * D_ + i]   = y[K_ + i];
    for (int i = 0; i < K_ * D_; ++i)     o_scales[nt * K_ * D_ + i]  = __expf(y[K_ + K_ * D_ + i]);
    for (int k = 0; k < K_; ++k)          o_corrs[nt * K_ + k]        = tanhf(y[K_ + 2 * K_ * D_ + k]);
    o_vlog[nt] = y[K_ + 2 * K_ * D_ + K_];
}

// ======================================================================
// Host launcher
// ======================================================================
static inline dim3 cdiv(int n, int b) { return dim3((unsigned)((n + b - 1) / b)); }

static inline void launch_gemm(int act, int acc,
                               const float* A, int lda, const float* B, int ldb,
                               const float* bias, float* C, int ldc,
                               int M, int Ntot, int K, hipStream_t s)
{
    // 4 waves/block, each wave covers TN=2 N-tiles -> 128 columns per block
    dim3 grid((unsigned)((Ntot + 127) / 128), (unsigned)(M / 16));
    dim3 blk(128);
    if (act == 0 && acc == 0)      gemm_wmma_f32<0, 0><<<grid, blk, 0, s>>>(A, lda, B, ldb, bias, C, ldc, M, Ntot, K);
    else if (act == 0 && acc == 1) gemm_wmma_f32<0, 1><<<grid, blk, 0, s>>>(A, lda, B, ldb, bias, C, ldc, M, Ntot, K);
    else                           gemm_wmma_f32<1, 0><<<grid, blk, 0, s>>>(A, lda, B, ldb, bias, C, ldc, M, Ntot, K);
}

extern "C" void kernel_launch(void* const* d_in, const int* in_sizes, int n_in,
                              void* d_out, int out_size, void* d_ws, size_t ws_size,
                              hipStream_t stream)
{
    const float* x       = (const float*)d_in[0];
    const float* x_canv  = (const float*)d_in[1];
    const float* start   = (const float*)d_in[2];
    const float* conv1_w = (const float*)d_in[3];
    const float* conv1_b = (const float*)d_in[4];
    const float* conv2_w = (const float*)d_in[5];
    const float* conv2_b = (const float*)d_in[6];
    const float* conv3_w = (const float*)d_in[7];
    const float* conv3_b = (const float*)d_in[8];
    const float* init_w  = (const float*)d_in[9];
    const float* init_b  = (const float*)d_in[10];
    const float* Uw      = (const float*)d_in[11];
    const float* Ub      = (const float*)d_in[12];
    const float* Ww      = (const float*)d_in[13];
    const float* Wb      = (const float*)d_in[14];
    const float* vw      = (const float*)d_in[15];
    const float* vb      = (const float*)d_in[16];
    const float* beta_w  = (const float*)d_in[17];
    const float* beta_b  = (const float*)d_in[18];
    const float* xemb_w  = (const float*)d_in[19];
    const float* xemb_b  = (const float*)d_in[20];
    const float* semb_w  = (const float*)d_in[21];
    const float* semb_b  = (const float*)d_in[22];
    const float* Wih     = (const float*)d_in[23];
    const float* Whh     = (const float*)d_in[24];
    const float* bih     = (const float*)d_in[25];
    const float* bhh     = (const float*)d_in[26];
    const float* comb_w  = (const float*)d_in[27];
    const float* comb_b  = (const float*)d_in[28];
    const float* out_w   = (const float*)d_in[29];
    const float* out_b   = (const float*)d_in[30];
    (void)in_sizes; (void)n_in; (void)out_size; (void)ws_size;

    // ---- workspace bump allocator (floats) ----
    size_t off = 0;
    float* wsf = (float*)d_ws;
    auto alloc = [&](size_t nelem) { float* p = wsf + off; off += nelem; return p; };
    float* xe      = alloc((size_t)N_ * T_ * 16);
    float* se      = alloc((size_t)N_ * 16);
    float* c1      = alloc((size_t)N_ * 16 * 784);
    float* c2      = alloc((size_t)N_ * 32 * 784);
    float* a       = alloc((size_t)N_ * L_ * FMAP_);
    float* amean   = alloc((size_t)N_ * FMAP_);
    float* h       = alloc((size_t)N_ * HID_);
    float* c       = alloc((size_t)N_ * HID_);
    float* Wa      = alloc((size_t)N_ * L_ * ATT_);
    float* Uh      = alloc((size_t)N_ * ATT_);
    float* betapre = alloc((size_t)N_ * FMAP_);
    float* z       = alloc((size_t)N_ * FMAP_);
    float* xz      = alloc((size_t)N_ * 80);
    float* gates   = alloc((size_t)N_ * 4 * HID_);
    float* u       = alloc((size_t)N_ * 608);
    float* y1      = alloc((size_t)N_ * COMB_);
    float* ys      = alloc((size_t)T_ * N_ * 128);

    const int B = 256;

    // ---- setup ----
    embed_kernel<<<cdiv(N_ * (T_ + 1) * 16, B), B, 0, stream>>>(x, start, xemb_w, xemb_b,
                                                                semb_w, semb_b, xe, se);
    conv1_kernel<<<cdiv(N_ * 16 * 784, B), B, 0, stream>>>(x_canv, conv1_w, conv1_b, c1);
    conv2_kernel<<<cdiv(N_ * 32 * 784, B), B, 0, stream>>>(c1, conv2_w, conv2_b, c2);
    conv3_kernel<<<cdiv(N_ * 64 * 196, B), B, 0, stream>>>(c2, conv3_w, conv3_b, a);
    amean_kernel<<<cdiv(N_ * FMAP_, B), B, 0, stream>>>(a, amean);
    init_state_kernel<<<cdiv(N_ * 2 * HID_, B), B, 0, stream>>>(amean, se, init_w, init_b, h, c);
    // W_a = a @ Ww^T + Wb : M = N*L = 25088, N = 128, K = 64
    launch_gemm(0, 0, a, FMAP_, Ww, FMAP_, Wb, Wa, ATT_, N_ * L_, ATT_, FMAP_, stream);

    // ---- 64-step scan ----
    for (int t = 0; t < T_; ++t) {
        // Uh = h @ Uw^T + Ub       (128 x 128, K=512)
        launch_gemm(0, 0, h, HID_, Uw, HID_, Ub, Uh, ATT_, N_, ATT_, HID_, stream);
        // betapre = h @ beta_w^T + beta_b  (128 x 64, K=512)
        launch_gemm(0, 0, h, HID_, beta_w, HID_, beta_b, betapre, FMAP_, N_, FMAP_, HID_, stream);
        // attention softmax + context
        attn_kernel<<<N_, 256, 0, stream>>>(Wa, Uh, vw, vb, a, betapre, z);
        // xz = [xe_t, z]
        pack_xz_kernel<<<cdiv(N_ * 80, B), B, 0, stream>>>(xe, z, xz, t);
        // gates = xz @ Wih^T + bih       (128 x 2048, K=80)
        launch_gemm(0, 0, xz, 80, Wih, 80, bih, gates, 4 * HID_, N_, 4 * HID_, 80, stream);
        // gates += h @ Whh^T + bhh       (128 x 2048, K=512)  <- dominant GEMM
        launch_gemm(0, 1, h, HID_, Whh, HID_, bhh, gates, 4 * HID_, N_, 4 * HID_, HID_, stream);
        // LSTM pointwise + pack u = [h, z, xt, se]
        lstm_kernel<<<cdiv(N_ * 608, B), B, 0, stream>>>(gates, c, h, z, xe, se, u, t);
        // y1 = tanh(u @ comb_w^T + comb_b)   (128 x 1024, K=608)
        launch_gemm(1, 0, u, 608, comb_w, 608, comb_b, y1, COMB_, N_, COMB_, 608, stream);
        // ys[t] = y1 @ out_w^T + out_b       (128 x 121, K=1024)
        launch_gemm(0, 0, y1, COMB_, out_w, COMB_, out_b,
                    ys + (size_t)t * N_ * 128, 128, N_, ODIM_, COMB_, stream);
    }

    // ---- output head transforms ----
    finalize_kernel<<<cdiv(N_ * T_, B), B, 0, stream>>>(ys, (float*)d_out);
}